// KnowledgeInteractionModule_16801912062748
// MI455X (gfx1250) — compile-verified
//
#include <hip/hip_runtime.h>
#include <hip/hip_bf16.h>
#include <math.h>

// CDNA5 / gfx1250, wave32. f32 WMMA fragments (ISA 7.12.2):
//   A (16x4 f32): 2 VGPRs. lane l<16: row M=l, {K=0,K=1}; lane 16+l: row M=l, {K=2,K=3}
//   B (4x16 f32): 2 VGPRs. lane l<16: col N=l, {K=0,K=1}; lane 16+l: col N=l, {K=2,K=3}
//   C/D (16x16 f32): 8 VGPRs. VGPR r: lanes 0-15 -> M=r, lanes 16-31 -> M=8+r; N = lane%16
typedef __attribute__((ext_vector_type(2))) float v2f;
typedef __attribute__((ext_vector_type(8))) float v8f;

__device__ __forceinline__ float gelu_exact(float x) {
    return 0.5f * x * (1.0f + erff(x * 0.70710678118654752f));
}

__device__ __forceinline__ float wave_reduce_sum(float v) {
    #pragma unroll
    for (int off = 16; off > 0; off >>= 1)
        v += __shfl_down(v, off, 32);   // wave32
    return v;
}

__device__ __forceinline__ v8f wmma4(v2f a, v2f b, v8f c) {
    return __builtin_amdgcn_wmma_f32_16x16x4_f32(false, a, false, b, (short)0, c, false, false);
}

// C[M,N] = alpha * (X[M,K] @ op(W)) + bias[n]
//   TRANS_B=true : op(W)[k,n] = W[n*ldw + k]   (X @ W^T, W row-major [N,K])
//   TRANS_B=false: op(W)[k,n] = W[k*ldw + n]   (X @ W,   W row-major [K,N])
// One wave computes a 32x64 output strip: per k-step, 2 A fragments x 4 B fragments
// feed 8 independent WMMA chains (6 b64 loads -> 8 WMMAs). The 8 chains fully hide
// the WMMA->WMMA RAW hazard and minimize VMEM issue pressure per matrix op.
template <bool TRANS_B>
__global__ void wmma_gemm_f32(const float* __restrict__ X, int ldx,
                              const float* __restrict__ W, int ldw,
                              float* __restrict__ C, int ldc,
                              const float* __restrict__ bias,
                              float alpha, int M, int N, int K)
{
    const int lane = threadIdx.x & 31;
    const int wv   = threadIdx.x >> 5;
    const int tilesN = N >> 6;                       // 64-wide strips
    const int tile = blockIdx.x * 8 + wv;            // 8 waves / block
    const int totalTiles = (M >> 5) * tilesN;        // 32-tall strips
    if (tile >= totalTiles) return;                  // wave-uniform: EXEC stays all-ones
    const int tm = (tile / tilesN) << 5;
    const int tn = (tile % tilesN) << 6;
    const int hg = lane >> 4;                        // K-pair select
    const int l  = lane & 15;

    v8f acc00 = {}, acc01 = {}, acc02 = {}, acc03 = {};
    v8f acc10 = {}, acc11 = {}, acc12 = {}, acc13 = {};
    const float* xrow0 = X + (size_t)(tm + l) * ldx + 2 * hg;
    const float* xrow1 = xrow0 + (size_t)16 * ldx;

    for (int k = 0; k < K; k += 4) {
        v2f a0, a1;
        a0.x = xrow0[k]; a0.y = xrow0[k + 1];
        a1.x = xrow1[k]; a1.y = xrow1[k + 1];
        __builtin_prefetch(xrow0 + k + 256, 0, 1);   // speculative: safe past end

        v2f b0, b1, b2, b3;
        if (TRANS_B) {
            const float* wp = W + (size_t)(tn + l) * ldw + k + 2 * hg;
            const size_t step = (size_t)16 * ldw;    // next 16-col group
            b0.x = wp[0];            b0.y = wp[1];
            b1.x = wp[step];         b1.y = wp[step + 1];
            b2.x = wp[2 * step];     b2.y = wp[2 * step + 1];
            b3.x = wp[3 * step];     b3.y = wp[3 * step + 1];
        } else {
            const float* wp = W + (size_t)(k + 2 * hg) * ldw + tn + l;
            b0.x = wp[0];   b0.y = wp[ldw];
            b1.x = wp[16];  b1.y = wp[ldw + 16];
            b2.x = wp[32];  b2.y = wp[ldw + 32];
            b3.x = wp[48];  b3.y = wp[ldw + 48];
        }
        acc00 = wmma4(a0, b0, acc00);
        acc01 = wmma4(a0, b1, acc01);
        acc02 = wmma4(a0, b2, acc02);
        acc03 = wmma4(a0, b3, acc03);
        acc10 = wmma4(a1, b0, acc10);
        acc11 = wmma4(a1, b1, acc11);
        acc12 = wmma4(a1, b2, acc12);
        acc13 = wmma4(a1, b3, acc13);
    }

    v8f accs[2][4] = {{acc00, acc01, acc02, acc03}, {acc10, acc11, acc12, acc13}};
    #pragma unroll
    for (int mt = 0; mt < 2; ++mt) {
        float* crow = C + (size_t)(tm + mt * 16 + 8 * hg) * ldc + tn + l;
        #pragma unroll
        for (int nt = 0; nt < 4; ++nt) {
            const int col = tn + nt * 16 + l;
            const float bv = bias ? bias[col] : 0.0f;
            #pragma unroll
            for (int r = 0; r < 8; ++r)
                crow[(size_t)r * ldc + nt * 16] = alpha * accs[mt][nt][r] + bv;
        }
    }
}

// scores[i,j] = sigmoid( sum_d gelu(A[i,d]+B[j,d]+b1[d]) * w2[d] + b2 )
// block = 256 threads = 8 waves; block handles (i, 8 consecutive j).
// float4 loads everywhere -> global_load_b128 / ds_load_b128.
__global__ void scores_kernel(const float* __restrict__ A,
                              const float* __restrict__ Bm,
                              const float* __restrict__ b1,
                              const float* __restrict__ w2,
                              const float* __restrict__ b2,
                              float* __restrict__ scores,
                              int N, int D)
{
    __shared__ float4 sA[256];                      // D/4 = 256 float4 = 4 KB
    const int i = blockIdx.x;
    const float4* Arow = (const float4*)(A + (size_t)i * D);
    const float4* b1v  = (const float4*)b1;
    const int D4 = D >> 2;
    for (int d = threadIdx.x; d < D4; d += blockDim.x) {
        float4 av = Arow[d], bv = b1v[d];
        sA[d] = make_float4(av.x + bv.x, av.y + bv.y, av.z + bv.z, av.w + bv.w);
    }
    __syncthreads();

    const int wv   = threadIdx.x >> 5;
    const int lane = threadIdx.x & 31;
    const int j    = blockIdx.y * 8 + wv;
    const float4* brow = (const float4*)(Bm + (size_t)j * D);
    const float4* w2v  = (const float4*)w2;
    float acc = 0.f;
    for (int q = lane; q < D4; q += 32) {
        float4 s = sA[q], b = brow[q], w = w2v[q];
        acc += gelu_exact(s.x + b.x) * w.x;
        acc += gelu_exact(s.y + b.y) * w.y;
        acc += gelu_exact(s.z + b.z) * w.z;
        acc += gelu_exact(s.w + b.w) * w.w;
    }
    acc = wave_reduce_sum(acc);
    if (lane == 0) {
        float s = acc + b2[0];
        scores[(size_t)i * N + j] = 1.0f / (1.0f + expf(-s));
    }
}

// Row softmax, block of 384 threads (12 waves) per row of length 384.
__global__ void softmax_rows(float* __restrict__ S)
{
    __shared__ float sred[12];
    float* row = S + (size_t)blockIdx.x * 384;
    const int t = threadIdx.x;
    const int wv = t >> 5, lane = t & 31;
    float v = row[t];

    float m = v;
    #pragma unroll
    for (int off = 16; off > 0; off >>= 1)
        m = fmaxf(m, __shfl_xor(m, off, 32));
    if (lane == 0) sred[wv] = m;
    __syncthreads();
    float gmax = sred[0];
    #pragma unroll
    for (int w = 1; w < 12; ++w) gmax = fmaxf(gmax, sred[w]);

    float e = expf(v - gmax);
    float s = e;
    #pragma unroll
    for (int off = 16; off > 0; off >>= 1)
        s += __shfl_xor(s, off, 32);
    __syncthreads();
    if (lane == 0) sred[wv] = s;
    __syncthreads();
    float gsum = 0.f;
    #pragma unroll
    for (int w = 0; w < 12; ++w) gsum += sred[w];
    row[t] = e / gsum;
}

// syn_in[0:2D] = mem rows 0,1 (contiguous); syn_in[2D+d] = mean_n attended[n,d]
__global__ void mean_concat(const float* __restrict__ attended,
                            const float* __restrict__ mem,
                            float* __restrict__ syn_in,
                            int N, int D)
{
    const int t = blockIdx.x * blockDim.x + threadIdx.x; // 0..3*D-1
    if (t < 2 * D) {
        syn_in[t] = mem[t];
    } else {
        const int d = t - 2 * D;
        float s = 0.f;
        for (int n = 0; n < N; ++n) s += attended[(size_t)n * D + d];
        syn_in[t] = s * (1.0f / (float)N);
    }
}

// y[n] = (gelu?)(dot(x, Wt[n,:]) + bias[n]); one wave per output; float4 loads.
__global__ void gemv_act(const float* __restrict__ x, const float* __restrict__ Wt,
                         const float* __restrict__ bias, float* __restrict__ y,
                         int Kdim, int Ndim, int apply_gelu)
{
    const int wv   = (blockIdx.x * blockDim.x + threadIdx.x) >> 5;
    const int lane = threadIdx.x & 31;
    if (wv >= Ndim) return;
    const float4* xv = (const float4*)x;
    const float4* wr = (const float4*)(Wt + (size_t)wv * Kdim);
    const int K4 = Kdim >> 2;
    float acc = 0.f;
    for (int q = lane; q < K4; q += 32) {
        float4 a = xv[q], b = wr[q];
        acc += a.x * b.x + a.y * b.y + a.z * b.z + a.w * b.w;
    }
    acc = wave_reduce_sum(acc);
    if (lane == 0) {
        float v = acc + bias[wv];
        y[wv] = apply_gelu ? gelu_exact(v) : v;
    }
}

extern "C" void kernel_launch(void* const* d_in, const int* in_sizes, int n_in,
                              void* d_out, int out_size, void* d_ws, size_t ws_size,
                              hipStream_t stream) {
    (void)in_sizes; (void)n_in; (void)out_size; (void)ws_size;
    const float* mem   = (const float*)d_in[0];
    const float* W1    = (const float*)d_in[1];
    const float* b1    = (const float*)d_in[2];
    const float* w2    = (const float*)d_in[3];
    const float* b2    = (const float*)d_in[4];
    const float* in_w  = (const float*)d_in[5];
    const float* in_b  = (const float*)d_in[6];
    const float* out_w = (const float*)d_in[7];
    const float* out_b = (const float*)d_in[8];
    const float* Ws1   = (const float*)d_in[9];
    const float* bs1   = (const float*)d_in[10];
    const float* Ws2   = (const float*)d_in[11];
    const float* bs2   = (const float*)d_in[12];

    const int N = 384, D = 1024, H = 8, hd = 128;

    float* ws   = (float*)d_ws;
    float* A    = ws;                      // N*D
    float* Bm   = A   + (size_t)N * D;     // N*D
    float* qkv  = Bm  + (size_t)N * D;     // N*3D
    float* S    = qkv + (size_t)N * 3 * D; // H*N*N
    float* att  = S   + (size_t)H * N * N; // N*D
    float* attd = att + (size_t)N * D;     // N*D
    float* syn  = attd+ (size_t)N * D;     // 3D
    float* hid  = syn + 3 * D;             // 2D

    float* scores = (float*)d_out;         // N*N
    float* nk     = scores + (size_t)N * N;// D

    dim3 blk(256);

    // A = mem @ W1[:, :D]^T ; B = mem @ W1[:, D:]^T   (b1 folded into scores kernel)
    // strips = (384/32) * (1024/64) = 12*16 = 192 -> 24 blocks of 8 waves
    wmma_gemm_f32<true><<<24, blk, 0, stream>>>(mem, D, W1,     2 * D, A,  D, nullptr, 1.f, N, D, D);
    wmma_gemm_f32<true><<<24, blk, 0, stream>>>(mem, D, W1 + D, 2 * D, Bm, D, nullptr, 1.f, N, D, D);
    // qkv = mem @ in_w^T + in_b : 12*48 = 576 strips -> 72 blocks
    wmma_gemm_f32<true><<<72, blk, 0, stream>>>(mem, D, in_w, D, qkv, 3 * D, in_b, 1.f, N, 3 * D, D);

    // pairwise analogy scores (dominant VALU/erf work)
    scores_kernel<<<dim3(N, N / 8), blk, 0, stream>>>(A, Bm, b1, w2, b2, scores, N, D);

    // per-head attention logits S_h = (q_h @ k_h^T) / sqrt(hd) : 12*6 = 72 strips -> 9 blocks
    const float scale = 1.0f / sqrtf((float)hd);
    for (int h = 0; h < H; ++h)
        wmma_gemm_f32<true><<<9, blk, 0, stream>>>(
            qkv + h * hd, 3 * D, qkv + D + h * hd, 3 * D,
            S + (size_t)h * N * N, N, nullptr, scale, N, N, hd);

    softmax_rows<<<H * N, 384, 0, stream>>>(S);

    // O_h = P_h @ v_h (non-transposed B) : 12*2 = 24 strips -> 3 blocks
    for (int h = 0; h < H; ++h)
        wmma_gemm_f32<false><<<3, blk, 0, stream>>>(
            S + (size_t)h * N * N, N, qkv + 2 * D + h * hd, 3 * D,
            att + h * hd, D, nullptr, 1.f, N, hd, N);

    // attended = att @ out_w^T + out_b : 24 blocks
    wmma_gemm_f32<true><<<24, blk, 0, stream>>>(att, D, out_w, D, attd, D, out_b, 1.f, N, D, D);

    // relations mean + concat(mem[0], mem[1], relations)
    mean_concat<<<(3 * D) / 256, 256, 0, stream>>>(attd, mem, syn, N, D);

    // synthesis MLP
    gemv_act<<<(2048 * 32) / 256, 256, 0, stream>>>(syn, Ws1, bs1, hid, 3 * D, 2 * D, 1);
    gemv_act<<<(1024 * 32) / 256, 256, 0, stream>>>(hid, Ws2, bs2, nk, 2 * D, D, 0);
}